// TransformerLayer_2705829396679
// MI455X (gfx1250) — compile-verified
//
#include <hip/hip_runtime.h>
#include <math.h>

#define B_   4
#define S_   2048
#define D_   256
#define H_   8
#define HD_  32
#define DFF_ 512

typedef __attribute__((ext_vector_type(2))) float v2f;
typedef __attribute__((ext_vector_type(4))) float v4f;
typedef __attribute__((ext_vector_type(8))) float v8f;

__device__ __forceinline__ v8f wmma_f32(v2f a, v2f b, v8f c) {
  // D = A(16x4 f32) x B(4x16 f32) + C(16x16 f32)
  return __builtin_amdgcn_wmma_f32_16x16x4_f32(false, a, false, b, (short)0, c,
                                               false, false);
}

// CDNA5 async global->LDS copies (ASYNCcnt-tracked), see CDNA5 ISA.
// Low 32 bits of a generic pointer to LDS == wave-relative LDS byte offset.
__device__ __forceinline__ void async_load_b128(const void* gptr, void* lptr) {
  unsigned lds = (unsigned)(uintptr_t)lptr;
  asm volatile("global_load_async_to_lds_b128 %0, %1, off"
               :: "v"(lds), "v"(gptr) : "memory");
}
__device__ __forceinline__ void async_load_b64(const void* gptr, void* lptr) {
  unsigned lds = (unsigned)(uintptr_t)lptr;
  asm volatile("global_load_async_to_lds_b64 %0, %1, off"
               :: "v"(lds), "v"(gptr) : "memory");
}
// Wait until at most 4 async ops outstanding (the 4 prefetches for the next tile).
__device__ __forceinline__ void wait_async_le4() {
  asm volatile("s_wait_asynccnt 0x4" ::: "memory");
}

enum { EPI_QKV = 0, EPI_BIAS_RES = 1, EPI_SILU = 2 };

// C[M,N] = X[M,K] @ W[N,K]^T (+ epilogue). Block tile 64x64, 256 threads,
// double-buffered async global->LDS pipeline over 32-wide K chunks.
template <int EPI, int K, int N>
__global__ __launch_bounds__(256) void gemm_kernel(
    const float* __restrict__ X, const float* __restrict__ W,
    const float* __restrict__ bias, const float* __restrict__ res,
    float* __restrict__ out) {
  __shared__ float Xs[2][64][36];  // +4 pad: 16B-aligned rows, bank-spread reads
  __shared__ float Ws[2][64][36];
  const int tid  = threadIdx.x;
  const int wave = tid >> 5, lane = tid & 31;
  const int half = lane >> 4, l16 = lane & 15;
  const int mw = wave & 3, nw = wave >> 2;  // wave -> (row16, col32) subtile
  const int m0 = blockIdx.x * 64, n0 = blockIdx.y * 64;
  const int lr = tid >> 3;            // 0..31 cooperative load row
  const int lc = (tid & 7) << 2;      // float4 column
  v8f acc0 = {}, acc1 = {};

  auto issue = [&](int k0, int buf) {  // 4 x B128 async per thread
    async_load_b128(&X[(size_t)(m0 + lr) * K + k0 + lc],      &Xs[buf][lr][lc]);
    async_load_b128(&X[(size_t)(m0 + lr + 32) * K + k0 + lc], &Xs[buf][lr + 32][lc]);
    async_load_b128(&W[(size_t)(n0 + lr) * K + k0 + lc],      &Ws[buf][lr][lc]);
    async_load_b128(&W[(size_t)(n0 + lr + 32) * K + k0 + lc], &Ws[buf][lr + 32][lc]);
  };

  constexpr int NIT = K / 32;
  issue(0, 0);
  for (int it = 0; it < NIT; ++it) {
    const int nk = (it + 1 < NIT) ? (it + 1) * 32 : it * 32;  // clamp: dup last
    issue(nk, (it + 1) & 1);
    wait_async_le4();   // current tile's 4 loads done; prefetch still in flight
    __syncthreads();    // -> all waves' loads for this tile complete
    const int cur = it & 1;
#pragma unroll
    for (int kk = 0; kk < 32; kk += 4) {
      // A frag: a[v] = X[m0+mw*16+l16][kk + 2*half + v]
      v2f a  = *(const v2f*)&Xs[cur][mw * 16 + l16][kk + 2 * half];
      // B frag: b[v] = W^T[k][n] = W[n][k], lane<->N, (group,v)<->K
      v2f b0 = *(const v2f*)&Ws[cur][nw * 32 + l16][kk + 2 * half];
      v2f b1 = *(const v2f*)&Ws[cur][nw * 32 + 16 + l16][kk + 2 * half];
      acc0 = wmma_f32(a, b0, acc0);
      acc1 = wmma_f32(a, b1, acc1);
    }
    __syncthreads();    // reads of buf `cur` done -> iter it+1 may overwrite it
  }

#pragma unroll
  for (int t = 0; t < 2; ++t) {
    v8f acc = (t == 0) ? acc0 : acc1;
#pragma unroll
    for (int r = 0; r < 8; ++r) {
      const int m = m0 + mw * 16 + r + 8 * half;       // C/D: M = r + 8*group
      const int n = n0 + nw * 32 + t * 16 + l16;       // C/D: N = lane&15
      float vv = acc[r] + bias[n];
      if constexpr (EPI == EPI_QKV) {
        // scatter qkv[m, n] -> {q|k|v}[b, h, s, d]  (each of size B*S*D)
        const int which = n >> 8;
        const int rem = n & 255;
        const int hh = rem >> 5, dd = rem & 31;
        const int bb = m >> 11, ss = m & 2047;
        out[(size_t)which * ((size_t)B_ * S_ * D_) +
            ((((size_t)bb * H_ + hh) * S_ + ss) * HD_ + dd)] = vv;
      } else if constexpr (EPI == EPI_BIAS_RES) {
        out[(size_t)m * N + n] = vv + res[(size_t)m * N + n];
      } else {  // EPI_SILU: x * sigmoid(x) via hardware v_exp_f32
        out[(size_t)m * N + n] = vv / (1.0f + __expf(-vv));
      }
    }
  }
}

// Flash attention with additive log-mask.  q/k/v: [B*H, S, HD].
// Block: 128 threads (4 waves, each owns 16 q rows). Grid: (S/64, B*H).
// K/V tiles double-buffered via async global->LDS; mask streamed non-temporal.
__global__ __launch_bounds__(128) void attn_kernel(
    const float* __restrict__ q, const float* __restrict__ k,
    const float* __restrict__ v, const float* __restrict__ mask,
    float* __restrict__ out) {
  __shared__ float Ks[2][16][34];
  __shared__ float Vs[2][16][34];
  __shared__ float Ps[4][16][18];  // per-wave P tile (C-layout -> A-layout)
  const int tid  = threadIdx.x;
  const int wave = tid >> 5, lane = tid & 31;
  const int half = lane >> 4, l16 = lane & 15;
  const int bh = blockIdx.y;
  const int b = bh >> 3, h = bh & 7;
  const int q0 = blockIdx.x * 64 + wave * 16;
  const float* qp = q + (size_t)bh * S_ * HD_;
  const float* kp = k + (size_t)bh * S_ * HD_;
  const float* vp = v + (size_t)bh * S_ * HD_;

  // Preload the wave's Q fragments for all 8 K-steps (HD=32 -> 8 x K4)
  v2f qf[8];
#pragma unroll
  for (int s = 0; s < 8; ++s)
    qf[s] = *(const v2f*)&qp[(size_t)(q0 + l16) * HD_ + 4 * s + 2 * half];

  v8f acc0 = {}, acc1 = {};
  float rmax[8], rsum[8];
#pragma unroll
  for (int r = 0; r < 8; ++r) { rmax[r] = -INFINITY; rsum[r] = 0.0f; }
  const float scale = 0.17677669529663687f;  // 1/sqrt(HD)

  // Each tile (16 rows x 32 floats = 2 KB) = 256 B64 chunks over 128 threads.
  const int c0 = tid << 1;  // chunks c0, c0+1 ; row = c>>4, col = (c&15)*2
  auto issue = [&](int kt, int buf) {  // 4 x B64 async per thread
#pragma unroll
    for (int j = 0; j < 2; ++j) {
      const int c = c0 + j;
      const int row = c >> 4, col = (c & 15) << 1;
      async_load_b64(&kp[(size_t)(kt * 16 + row) * HD_ + col], &Ks[buf][row][col]);
      async_load_b64(&vp[(size_t)(kt * 16 + row) * HD_ + col], &Vs[buf][row][col]);
    }
  };

  constexpr int NKT = S_ / 16;
  issue(0, 0);
  for (int kt = 0; kt < NKT; ++kt) {
    const int nk = (kt + 1 < NKT) ? kt + 1 : kt;  // clamp: dup last
    issue(nk, (kt + 1) & 1);
    wait_async_le4();
    __syncthreads();
    const int cur = kt & 1;

    // scores(16x16) = Q(16x32) @ K^T(32x16): b[v] = K[n][kdim]
    v8f sc = {};
#pragma unroll
    for (int s = 0; s < 8; ++s) {
      v2f bb = *(const v2f*)&Ks[cur][l16][4 * s + 2 * half];
      sc = wmma_f32(qf[s], bb, sc);
    }

    // mask + online softmax; row m = r + 8*half lives in a 16-lane half-wave.
    // mask is clamped to [1e-15, inf) -> always a normal float, so the
    // hardware v_log_f32 / v_exp_f32 paths are safe and cheap.
    float alpha[8];
#pragma unroll
    for (int r = 0; r < 8; ++r) {
      const int qrow = q0 + r + 8 * half;
      const float mval = __builtin_nontemporal_load(
          &mask[((size_t)b * S_ + qrow) * S_ + kt * 16 + l16]);
      const float val = sc[r] * scale + __logf(fmaxf(mval, 1e-15f));
      float mx = val;
      mx = fmaxf(mx, __shfl_xor(mx, 1, 32));
      mx = fmaxf(mx, __shfl_xor(mx, 2, 32));
      mx = fmaxf(mx, __shfl_xor(mx, 4, 32));
      mx = fmaxf(mx, __shfl_xor(mx, 8, 32));
      const float nm = fmaxf(rmax[r], mx);
      alpha[r] = __expf(rmax[r] - nm);
      rmax[r] = nm;
      const float pe = __expf(val - nm);
      float ps = pe;
      ps += __shfl_xor(ps, 1, 32);
      ps += __shfl_xor(ps, 2, 32);
      ps += __shfl_xor(ps, 4, 32);
      ps += __shfl_xor(ps, 8, 32);
      rsum[r] = rsum[r] * alpha[r] + ps;
      acc0[r] *= alpha[r];
      acc1[r] *= alpha[r];
      Ps[wave][r + 8 * half][l16] = pe;  // P in row-major for A-frag reload
    }

    // O(16x32) += P(16x16) @ V(16x32)
#pragma unroll
    for (int s = 0; s < 4; ++s) {
      v2f a = *(const v2f*)&Ps[wave][l16][4 * s + 2 * half];
      const int kr = 4 * s + 2 * half;
      v2f b0, b1;
      b0[0] = Vs[cur][kr][l16];      b0[1] = Vs[cur][kr + 1][l16];
      b1[0] = Vs[cur][kr][16 + l16]; b1[1] = Vs[cur][kr + 1][16 + l16];
      acc0 = wmma_f32(a, b0, acc0);
      acc1 = wmma_f32(a, b1, acc1);
    }
    __syncthreads();  // reads of buf `cur` done -> next iter may overwrite it
  }

  // write attn output back in [B, S, D] layout (head-interleaved)
#pragma unroll
  for (int r = 0; r < 8; ++r) {
    const int qrow = q0 + r + 8 * half;
    const float inv = 1.0f / rsum[r];
    const size_t base = ((size_t)b * S_ + qrow) * D_ + h * HD_;
    out[base + l16]      = acc0[r] * inv;
    out[base + 16 + l16] = acc1[r] * inv;
  }
}

// LayerNorm over rows of 256 floats; one wave per row, 8 rows per block.
__global__ __launch_bounds__(256) void ln_kernel(
    const float* __restrict__ in, const float* __restrict__ gamma,
    const float* __restrict__ beta, float* __restrict__ out) {
  const int wave = threadIdx.x >> 5, lane = threadIdx.x & 31;
  const int row = blockIdx.x * 8 + wave;
  const float* rp = in + (size_t)row * D_;
  float x[8];
  v4f a0 = *(const v4f*)&rp[lane * 8];
  v4f a1 = *(const v4f*)&rp[lane * 8 + 4];
#pragma unroll
  for (int i = 0; i < 4; ++i) { x[i] = a0[i]; x[i + 4] = a1[i]; }
  float sum = 0.0f;
#pragma unroll
  for (int i = 0; i < 8; ++i) sum += x[i];
  sum += __shfl_xor(sum, 16, 32);
  sum += __shfl_xor(sum, 8, 32);
  sum += __shfl_xor(sum, 4, 32);
  sum += __shfl_xor(sum, 2, 32);
  sum += __shfl_xor(sum, 1, 32);
  const float mu = sum * (1.0f / 256.0f);
  float sq = 0.0f;
#pragma unroll
  for (int i = 0; i < 8; ++i) { const float d = x[i] - mu; sq += d * d; }
  sq += __shfl_xor(sq, 16, 32);
  sq += __shfl_xor(sq, 8, 32);
  sq += __shfl_xor(sq, 4, 32);
  sq += __shfl_xor(sq, 2, 32);
  sq += __shfl_xor(sq, 1, 32);
  const float rs = rsqrtf(sq * (1.0f / 256.0f) + 1e-5f);
#pragma unroll
  for (int i = 0; i < 8; ++i) {
    const int col = lane * 8 + i;
    out[(size_t)row * D_ + col] = (x[i] - mu) * rs * gamma[col] + beta[col];
  }
}

extern "C" void kernel_launch(void* const* d_in, const int* in_sizes, int n_in,
                              void* d_out, int out_size, void* d_ws,
                              size_t ws_size, hipStream_t stream) {
  (void)in_sizes; (void)n_in; (void)out_size; (void)ws_size;
  const float* tokens = (const float*)d_in[0];
  const float* mask   = (const float*)d_in[1];
  const float* Wqkv   = (const float*)d_in[2];
  const float* bqkv   = (const float*)d_in[3];
  const float* Wo     = (const float*)d_in[4];
  const float* bo     = (const float*)d_in[5];
  const float* ln1g   = (const float*)d_in[6];
  const float* ln1b   = (const float*)d_in[7];
  const float* ln2g   = (const float*)d_in[8];
  const float* ln2b   = (const float*)d_in[9];
  const float* W1     = (const float*)d_in[10];
  const float* b1     = (const float*)d_in[11];
  const float* W2     = (const float*)d_in[12];
  const float* b2     = (const float*)d_in[13];
  float* outp = (float*)d_out;

  // workspace layout (units of U = B*S*D floats = 8 MB), with reuse:
  //   [0]=q (later x), [1]=k (later y2), [2]=v (later y1), [3..4]=attn/h
  float* fws = (float*)d_ws;
  const size_t U = (size_t)B_ * S_ * D_;
  float* qb    = fws;
  float* kb    = fws + U;
  float* vb    = fws + 2 * U;
  float* attnb = fws + 3 * U;
  float* y1    = fws + 2 * U;  // v dead after attention
  float* xb    = fws;          // q dead after attention
  float* hb    = fws + 3 * U;  // attn dead after Wo proj (2 units: DFF=2D)
  float* y2    = fws + U;      // k dead after attention

  dim3 blk(256);
  // 1) QKV projection + scatter to per-head layout
  gemm_kernel<EPI_QKV, 256, 768><<<dim3(128, 12), blk, 0, stream>>>(
      tokens, Wqkv, bqkv, nullptr, qb);
  // 2) masked flash attention
  attn_kernel<<<dim3(32, 32), dim3(128), 0, stream>>>(qb, kb, vb, mask, attnb);
  // 3) output projection + bias + residual(tokens)
  gemm_kernel<EPI_BIAS_RES, 256, 256><<<dim3(128, 4), blk, 0, stream>>>(
      attnb, Wo, bo, tokens, y1);
  // 4) LN1
  ln_kernel<<<dim3(1024), blk, 0, stream>>>(y1, ln1g, ln1b, xb);
  // 5) MLP up-proj + SiLU
  gemm_kernel<EPI_SILU, 256, 512><<<dim3(128, 8), blk, 0, stream>>>(
      xb, W1, b1, nullptr, hb);
  // 6) MLP down-proj + bias + residual(x)
  gemm_kernel<EPI_BIAS_RES, 512, 256><<<dim3(128, 4), blk, 0, stream>>>(
      hb, W2, b2, xb, y2);
  // 7) LN2 -> final output
  ln_kernel<<<dim3(1024), blk, 0, stream>>>(y2, ln2g, ln2b, outp);
}